// GraphSAGE_85126251807613
// MI455X (gfx1250) — compile-verified
//
#include <hip/hip_runtime.h>
#include <cstdint>
#include <cstddef>

#define N_NODES 50000
#define N_EDGES 800000
#define D 128

typedef float v2f __attribute__((ext_vector_type(2)));
typedef float v8f __attribute__((ext_vector_type(8)));

// ---------------------------------------------------------------------------
// Degree accumulation: one thread per edge, atomic add into deg[dst].
// ---------------------------------------------------------------------------
__global__ void sage_deg_kernel(const long long* __restrict__ dst,
                                float* __restrict__ deg, int nEdges) {
  int e = blockIdx.x * blockDim.x + threadIdx.x;
  if (e < nEdges) {
    atomicAdd(&deg[(int)dst[e]], 1.0f);
  }
}

// ---------------------------------------------------------------------------
// Scatter: msg[dst] += h[src].  32 threads per edge, 4 floats per thread
// (float4 gather + 4x global_atomic_add_f32).
// ---------------------------------------------------------------------------
__global__ void sage_scatter_kernel(const float* __restrict__ h,
                                    const long long* __restrict__ src,
                                    const long long* __restrict__ dst,
                                    float* __restrict__ msg, int nEdges) {
  int t = blockIdx.x * blockDim.x + threadIdx.x;
  int e = t >> 5;
  if (e >= nEdges) return;
  int c4 = (t & 31) * 4;
  const size_t soff = (size_t)src[e] * D + c4;
  const size_t doff = (size_t)dst[e] * D + c4;
  const float4 v = *reinterpret_cast<const float4*>(h + soff);
  float* p = msg + doff;
  atomicAdd(p + 0, v.x);
  atomicAdd(p + 1, v.y);
  atomicAdd(p + 2, v.z);
  atomicAdd(p + 3, v.w);
}

// ---------------------------------------------------------------------------
// Fused SAGEConv GEMM:
//   hout = relu?( (msg * 1/max(deg,1)) @ w_nbr + hin @ w_root + bias )
// One wave per 16x16 output tile via V_WMMA_F32_16X16X4_F32.
// Block = 256 threads = 8 waves covering the 8 column tiles (D=128).
// grid.x = N_NODES/16 = 3125 (exact -> EXEC all ones everywhere).
// ---------------------------------------------------------------------------
__global__ void sage_gemm_kernel(const float* __restrict__ hin,
                                 const float* __restrict__ msg,
                                 const float* __restrict__ deg,
                                 const float* __restrict__ w_nbr,
                                 const float* __restrict__ w_root,
                                 const float* __restrict__ bias,
                                 float* __restrict__ hout,
                                 int relu) {
  const int lane = threadIdx.x & 31;
  const int wave = threadIdx.x >> 5;          // 0..7 -> column tile
  const int row0 = blockIdx.x * 16;
  const int col0 = wave * 16;
  const int m    = lane & 15;                 // A row / BCD column within tile
  const int half = lane >> 4;                 // 0 or 1 -> K sub-pair select
  const int row  = row0 + m;

  const float scale = 1.0f / fmaxf(deg[row], 1.0f);

  // C init = bias broadcast along rows (each acc element shares column m).
  v8f acc;
  const float bv = bias[col0 + m];
#pragma unroll
  for (int r = 0; r < 8; ++r) acc[r] = bv;

  const float* arow_msg = msg  + (size_t)row * D + 2 * half;
  const float* arow_h   = hin  + (size_t)row * D + 2 * half;
  const float* bcol_n   = w_nbr  + (size_t)(2 * half) * D + col0 + m;
  const float* bcol_r   = w_root + (size_t)(2 * half) * D + col0 + m;

  // Pass 1: mean-aggregated neighbors  @ w_nbr  (scale folded into A loads)
#pragma unroll 8
  for (int k = 0; k < D; k += 4) {
    const float2 av = *reinterpret_cast<const float2*>(arow_msg + k);
    v2f a, b;
    a.x = av.x * scale;
    a.y = av.y * scale;
    b.x = bcol_n[(size_t)k * D];
    b.y = bcol_n[(size_t)(k + 1) * D];
    acc = __builtin_amdgcn_wmma_f32_16x16x4_f32(
        /*neg_a=*/false, a, /*neg_b=*/false, b,
        /*c_mod=*/(short)0, acc, /*reuse_a=*/false, /*reuse_b=*/false);
  }

  // Pass 2: root features @ w_root, same accumulator
#pragma unroll 8
  for (int k = 0; k < D; k += 4) {
    const float2 av = *reinterpret_cast<const float2*>(arow_h + k);
    v2f a, b;
    a.x = av.x;
    a.y = av.y;
    b.x = bcol_r[(size_t)k * D];
    b.y = bcol_r[(size_t)(k + 1) * D];
    acc = __builtin_amdgcn_wmma_f32_16x16x4_f32(
        false, a, false, b, (short)0, acc, false, false);
  }

  // D layout: VGPR r -> row (half*8 + r), column m.
  const int mbase = row0 + half * 8;
#pragma unroll
  for (int r = 0; r < 8; ++r) {
    float v = acc[r];
    if (relu) v = fmaxf(v, 0.0f);
    hout[(size_t)(mbase + r) * D + col0 + m] = v;
  }
}

// ---------------------------------------------------------------------------
// Classifier: out[n] = dot(h[n,:], w_cls) + b_cls.  One wave per node,
// 4 floats per lane, wave32 shuffle reduction.
// ---------------------------------------------------------------------------
__global__ void sage_cls_kernel(const float* __restrict__ h,
                                const float* __restrict__ w,
                                const float* __restrict__ b,
                                float* __restrict__ out) {
  const int lane = threadIdx.x & 31;
  const int wv   = threadIdx.x >> 5;
  const int node = blockIdx.x * 8 + wv;
  if (node >= N_NODES) return;
  const float4 hv = *reinterpret_cast<const float4*>(h + (size_t)node * D + lane * 4);
  const float4 wc = *reinterpret_cast<const float4*>(w + lane * 4);
  float s = hv.x * wc.x + hv.y * wc.y + hv.z * wc.z + hv.w * wc.w;
#pragma unroll
  for (int off = 16; off; off >>= 1) s += __shfl_down(s, off, 32);
  if (lane == 0) out[node] = s + b[0];
}

// ---------------------------------------------------------------------------
extern "C" void kernel_launch(void* const* d_in, const int* in_sizes, int n_in,
                              void* d_out, int out_size, void* d_ws, size_t ws_size,
                              hipStream_t stream) {
  const float*     x      = (const float*)d_in[0];
  const long long* edge   = (const long long*)d_in[1];   // int64 [2, E]
  const float* w_nbr[3]  = {(const float*)d_in[2], (const float*)d_in[5], (const float*)d_in[8]};
  const float* w_root[3] = {(const float*)d_in[3], (const float*)d_in[6], (const float*)d_in[9]};
  const float* bias[3]   = {(const float*)d_in[4], (const float*)d_in[7], (const float*)d_in[10]};
  const float* w_cls     = (const float*)d_in[11];
  const float* b_cls     = (const float*)d_in[12];
  float* out = (float*)d_out;

  const long long* src = edge;            // edge_index[0]
  const long long* dst = edge + N_EDGES;  // edge_index[1]

  // Workspace layout (256B aligned slices)
  char* ws = (char*)d_ws;
  size_t off = 0;
  auto take = [&](size_t bytes) -> char* {
    char* p = ws + off;
    off += (bytes + 255) & ~(size_t)255;
    return p;
  };
  const size_t hbytes = (size_t)N_NODES * D * sizeof(float);
  float* deg = (float*)take((size_t)N_NODES * sizeof(float));
  float* h0  = (float*)take(hbytes);
  float* h1  = (float*)take(hbytes);
  float* msg = (float*)take(hbytes);
  (void)ws_size; (void)in_sizes; (void)n_in; (void)out_size;

  const int degBlocks     = (N_EDGES + 255) / 256;
  const int scatterBlocks = (int)(((size_t)N_EDGES * 32 + 255) / 256);
  const int gemmBlocks    = N_NODES / 16;            // 3125 exact
  const int clsBlocks     = (N_NODES + 7) / 8;

  // Degrees (graph-invariant across layers)
  hipMemsetAsync(deg, 0, (size_t)N_NODES * sizeof(float), stream);
  sage_deg_kernel<<<degBlocks, 256, 0, stream>>>(dst, deg, N_EDGES);

  const float* hin = x;
  float* houts[3] = {h0, h1, h0};
  for (int l = 0; l < 3; ++l) {
    hipMemsetAsync(msg, 0, hbytes, stream);
    sage_scatter_kernel<<<scatterBlocks, 256, 0, stream>>>(hin, src, dst, msg, N_EDGES);
    sage_gemm_kernel<<<gemmBlocks, 256, 0, stream>>>(
        hin, msg, deg, w_nbr[l], w_root[l], bias[l], houts[l], /*relu=*/(l < 2) ? 1 : 0);
    hin = houts[l];
  }

  sage_cls_kernel<<<clsBlocks, 256, 0, stream>>>(hin, w_cls, b_cls, out);
}